// YoloV3_28260884807846
// MI455X (gfx1250) — compile-verified
//
#include <hip/hip_runtime.h>
#include <hip/hip_bf16.h>
#include <stdint.h>

// ---------------- problem constants (from reference) ----------------
#define B_IMG   64
#define NBOX    22743          // (19^2 + 38^2 + 76^2) * 3
#define OFF1    1083           // 19*19*3
#define OFF2    5415           // OFF1 + 38*38*3
#define NCLS    2
#define KC      512            // K_CAND
#define MAXPC   100
#define MAXT    100
#define IOU_T   0.5f
#define SCORE_T 0.5f

typedef __attribute__((ext_vector_type(2))) float        v2f;
typedef __attribute__((ext_vector_type(8))) float        v8f;
typedef __attribute__((ext_vector_type(4))) unsigned int v4u;
typedef __attribute__((ext_vector_type(8))) int          v8i;
typedef __attribute__((ext_vector_type(4))) int          v4i;

__device__ __forceinline__ float sigm(float x) { return 1.0f / (1.0f + __expf(-x)); }

// ordered-float key: descending value, ascending index tie-break (== lax.top_k)
__device__ __forceinline__ unsigned long long packKey(float v, unsigned idx) {
    unsigned u = __float_as_uint(v);
    u = (u & 0x80000000u) ? ~u : (u | 0x80000000u);
    return ((unsigned long long)u << 32) | (0xFFFFFFFFu - idx);
}
__device__ __forceinline__ float keyVal(unsigned long long k) {
    unsigned u = (unsigned)(k >> 32);
    return __uint_as_float((u & 0x80000000u) ? (u ^ 0x80000000u) : ~u);
}
__device__ __forceinline__ unsigned keyIdx(unsigned long long k) {
    return 0xFFFFFFFFu - (unsigned)(k & 0xFFFFFFFFu);
}

// ---------------- 1) decode: pred -> boxes[B][N][4], score planes [B][C][N] ----
__global__ __launch_bounds__(256) void decode_kernel(
    const float* __restrict__ p1, const float* __restrict__ p2,
    const float* __restrict__ p3, const float* __restrict__ anch,
    float* __restrict__ boxes, float* __restrict__ planes) {
    long long idx = (long long)blockIdx.x * blockDim.x + threadIdx.x;
    if (idx >= (long long)B_IMG * NBOX) return;
    int b = (int)(idx / NBOX);
    int n = (int)(idx % NBOX);

    int g, aoff, local;
    const float* p;
    if (n < OFF1)      { g = 19; aoff = 6; local = n;        p = p1 + ((size_t)b * OFF1          + local) * 7; }
    else if (n < OFF2) { g = 38; aoff = 3; local = n - OFF1; p = p2 + ((size_t)b * (OFF2 - OFF1) + local) * 7; }
    else               { g = 76; aoff = 0; local = n - OFF2; p = p3 + ((size_t)b * (NBOX - OFF2) + local) * 7; }

    int a    = local % 3;
    int cell = local / 3;
    int gj   = cell % g;     // x (grid[...,0] = gx = column)
    int gi   = cell / g;     // y

    float invg = 1.0f / (float)g;
    float cx = (sigm(p[0]) + (float)gj) * invg;
    float cy = (sigm(p[1]) + (float)gi) * invg;
    float w  = __expf(p[2]) * anch[(aoff + a) * 2 + 0];
    float h  = __expf(p[3]) * anch[(aoff + a) * 2 + 1];
    float x1 = cx - 0.5f * w, y1 = cy - 0.5f * h;

    size_t bo = ((size_t)b * NBOX + n) * 4;
    boxes[bo + 0] = x1; boxes[bo + 1] = y1;
    boxes[bo + 2] = x1 + w; boxes[bo + 3] = y1 + h;

    float obj = sigm(p[4]);
    planes[((size_t)b * NCLS + 0) * NBOX + n] = obj * sigm(p[5]);
    planes[((size_t)b * NCLS + 1) * NBOX + n] = obj * sigm(p[6]);
}

// ---------------- 2) top-512 per (b,c): TDM-staged scores + iterative argmax ----
__global__ __launch_bounds__(1024) void topk_kernel(
    const float* __restrict__ planes, const float* __restrict__ boxes,
    float* __restrict__ cand_boxes, float* __restrict__ cand_sc) {
    __shared__ float              s_sc[NBOX];      // 91 KB of the 320 KB/WGP LDS
    __shared__ unsigned long long kb[1024];
    __shared__ unsigned           s_idx[KC];
    __shared__ float              s_val[KC];

    const int bc = blockIdx.x;           // b*2 + c
    const int b  = bc >> 1;
    const unsigned tid = threadIdx.x;
    const float* plane = planes + (size_t)bc * NBOX;

    if (tid == 0) {
        // Tensor DMA Descriptor (D#): 1-D tile of NBOX fp32 -> LDS (CDNA5 ISA)
        unsigned lds_lo = (unsigned)(unsigned long long)(uintptr_t)&s_sc[0];
        unsigned long long ga = (unsigned long long)(uintptr_t)plane;
        v4u g0;
        g0.x = 1u;                                           // count=1 (valid descriptor)
        g0.y = lds_lo;                                       // lds_addr
        g0.z = (unsigned)(ga & 0xFFFFFFFFull);               // global_addr[31:0]
        g0.w = (unsigned)((ga >> 32) & 0x01FFFFFFull) | (2u << 30); // addr[56:32] | type=2
        v8i g1;
        g1[0] = (int)(2u << 16);                 // data_size=4B; no multicast/pad/iterate
        g1[1] = (int)((NBOX & 0xFFFFu) << 16);   // tensor_dim0[15:0] (NBOX < 2^16)
        g1[2] = (int)((NBOX >> 16) | (1u << 16));// tensor_dim0[31:16] | tensor_dim1=1
        g1[3] = (int)((NBOX & 0xFFFFu) << 16);   // tile_dim0 = NBOX
        g1[4] = 1;                               // tile_dim1 = 1, tile_dim2 = 0
        g1[5] = NBOX;                            // tensor_dim0_stride[31:0]
        g1[6] = 0; g1[7] = 0;
        v4i gz4 = {0, 0, 0, 0};
        v8i gz8 = {0, 0, 0, 0, 0, 0, 0, 0};
        // 6-arg toolchain form: (g0, g1, g2, g3, extra-group, cpol)
        __builtin_amdgcn_tensor_load_to_lds(g0, g1, gz4, gz4, gz8, 0);
        __builtin_amdgcn_s_wait_tensorcnt(0);
    }
    __syncthreads();

    for (int r = 0; r < KC; ++r) {
        unsigned long long best = 0ull;
        for (int n = tid; n < NBOX; n += 1024) {
            unsigned long long k = packKey(s_sc[n], (unsigned)n);
            best = (k > best) ? k : best;
        }
        kb[tid] = best;
        __syncthreads();
        for (int s = 512; s > 0; s >>= 1) {
            if (tid < (unsigned)s) {
                unsigned long long o = kb[tid + s];
                if (o > kb[tid]) kb[tid] = o;
            }
            __syncthreads();
        }
        if (tid == 0) {
            unsigned long long k = kb[0];
            unsigned n = keyIdx(k);
            s_idx[r] = n;
            s_val[r] = keyVal(k);
            s_sc[n]  = -3.0e38f;                 // remove winner
        }
        __syncthreads();
    }

    if (tid < KC) {
        unsigned n  = s_idx[tid];
        size_t src  = ((size_t)b * NBOX + n) * 4;
        size_t dst  = ((size_t)bc * KC + tid) * 4;
        cand_boxes[dst + 0] = boxes[src + 0];
        cand_boxes[dst + 1] = boxes[src + 1];
        cand_boxes[dst + 2] = boxes[src + 2];
        cand_boxes[dst + 3] = boxes[src + 3];
        cand_sc[(size_t)bc * KC + tid] = s_val[tid];
    }
}

// ---------------- 3) greedy NMS per (b,c): one wave32, WMMA-blocked suppression ----
__global__ __launch_bounds__(32) void nms_kernel(
    const float* __restrict__ cand_boxes, const float* __restrict__ cand_sc,
    float* __restrict__ fs) {
    __shared__ float bx[KC][4];
    __shared__ float ar[KC];
    __shared__ float sc[KC];
    __shared__ float kp[KC];      // keep flag as f32 0/1 (WMMA B operand)
    __shared__ float sup[16];

    const int bc = blockIdx.x;
    const int lane = threadIdx.x;

    for (int k = lane; k < KC; k += 32) {
        const float* p = cand_boxes + ((size_t)bc * KC + k) * 4;
        float x1 = p[0], y1 = p[1], x2 = p[2], y2 = p[3];
        bx[k][0] = x1; bx[k][1] = y1; bx[k][2] = x2; bx[k][3] = y2;
        ar[k] = fmaxf(x2 - x1, 0.f) * fmaxf(y2 - y1, 0.f);
        sc[k] = cand_sc[(size_t)bc * KC + k];
        kp[k] = 0.f;
    }
    __syncthreads();

    auto iouf = [&](int i, int j) -> float {
        float iw = fminf(bx[i][2], bx[j][2]) - fmaxf(bx[i][0], bx[j][0]);
        float ih = fminf(bx[i][3], bx[j][3]) - fmaxf(bx[i][1], bx[j][1]);
        iw = fmaxf(iw, 0.f); ih = fmaxf(ih, 0.f);
        float inter = iw * ih;
        float uni = ar[i] + ar[j] - inter;
        return inter / fmaxf(uni, 1e-9f);
    };

    for (int t = 0; t < KC / 16; ++t) {
        // suppressed_prev[row] = sum_{j in decided tiles} (iou(row,j)>T) * keep[j]
        // 16xK mask-matvec on the matrix pipe: V_WMMA_F32_16X16X4_F32.
        // A layout (ISA 7.12.2): lanes 0-15 hold K=0,1 ; lanes 16-31 hold K=2,3 (M = lane&15).
        v8f acc = {0.f, 0.f, 0.f, 0.f, 0.f, 0.f, 0.f, 0.f};
        int m = lane & 15;
        int i = t * 16 + m;
        for (int j0 = 0; j0 < t * 16; j0 += 4) {
            int kk = (lane < 16) ? 0 : 2;
            float a0 = (iouf(i, j0 + kk)     > IOU_T) ? 1.f : 0.f;
            float a1 = (iouf(i, j0 + kk + 1) > IOU_T) ? 1.f : 0.f;
            v2f A = {a0, a1};
            v2f Bv = {kp[j0 + kk], kp[j0 + kk + 1]};   // keep flags broadcast across N
            acc = __builtin_amdgcn_wmma_f32_16x16x4_f32(
                false, A, false, Bv, (short)0, acc, false, false);
        }
        // D layout: VGPR r holds M=r (lanes 0-15, N=lane) and M=r+8 (lanes 16-31).
        // Column N=0 lives in lane 0 (M=0..7) and lane 16 (M=8..15).
        if (lane == 0)  for (int r = 0; r < 8; ++r) sup[r]     = acc[r];
        if (lane == 16) for (int r = 0; r < 8; ++r) sup[8 + r] = acc[r];
        __syncthreads();
        // intra-tile sequential dependency (true serial part of greedy NMS)
        if (lane == 0) {
            for (int m2 = 0; m2 < 16; ++m2) {
                int i2 = t * 16 + m2;
                bool suppressed = sup[m2] > 0.f;
                for (int mm = 0; mm < m2 && !suppressed; ++mm)
                    if (kp[t * 16 + mm] > 0.f && iouf(i2, t * 16 + mm) > IOU_T)
                        suppressed = true;
                kp[i2] = ((sc[i2] > SCORE_T) && !suppressed) ? 1.f : 0.f;
            }
        }
        __syncthreads();
    }

    // keep &= cumsum(keep) <= MAX_PER_CLASS  (cumsum over pre-cap keep)
    if (lane == 0) {
        int cnt = 0;
        for (int k = 0; k < KC; ++k)
            if (kp[k] > 0.f) { if (++cnt > MAXPC) kp[k] = 0.f; }
    }
    __syncthreads();

    for (int k = lane; k < KC; k += 32)
        fs[(size_t)bc * KC + k] = (kp[k] > 0.f) ? sc[k] : -1.0f;
}

// ---------------- 4) per-image merge: bitonic top-100 over 1024 candidates ----
__global__ __launch_bounds__(1024) void final_kernel(
    const float* __restrict__ fs, const float* __restrict__ cand_boxes,
    float* __restrict__ out) {
    __shared__ unsigned long long keys[1024];
    const int b = blockIdx.x;
    const unsigned tid = threadIdx.x;

    keys[tid] = packKey(fs[(size_t)b * 1024 + tid], tid);  // flat idx = c*512 + k
    __syncthreads();

    for (int k2 = 2; k2 <= 1024; k2 <<= 1) {
        for (int j = k2 >> 1; j > 0; j >>= 1) {
            unsigned ixj = tid ^ (unsigned)j;
            if (ixj > tid) {
                unsigned long long a = keys[tid], c = keys[ixj];
                bool descBlock = ((tid & (unsigned)k2) == 0);
                bool doSwap = descBlock ? (a < c) : (a > c);
                if (doSwap) { keys[tid] = c; keys[ixj] = a; }
            }
            __syncthreads();
        }
    }

    const size_t OB = 0;
    const size_t OS = (size_t)B_IMG * MAXT * 4;            // 25600
    const size_t OC = OS + (size_t)B_IMG * MAXT;           // 32000
    const size_t ON = OC + (size_t)B_IMG * MAXT;           // 38400

    if (tid < MAXT) {
        unsigned long long k = keys[tid];
        float v = keyVal(k);
        unsigned f = keyIdx(k);
        bool kept = v > 0.f;
        unsigned c = f >> 9, kk = f & 511u;
        const float* p = cand_boxes + (((size_t)b * NCLS + c) * KC + kk) * 4;
        size_t ob = OB + ((size_t)b * MAXT + tid) * 4;
        #pragma unroll
        for (int d = 0; d < 4; ++d)
            out[ob + d] = kept ? fminf(fmaxf(p[d], 0.f), 1.f) : 0.f;
        out[OS + (size_t)b * MAXT + tid] = kept ? v : 0.f;
        out[OC + (size_t)b * MAXT + tid] = kept ? (float)c : 0.f;
    }
    if (tid == 0) {
        int cnt = 0;
        for (int r = 0; r < MAXT; ++r)
            if (keyVal(keys[r]) > 0.f) ++cnt;
        out[ON + b] = (float)cnt;
    }
}

// ---------------- launcher ----------------
extern "C" void kernel_launch(void* const* d_in, const int* in_sizes, int n_in,
                              void* d_out, int out_size, void* d_ws, size_t ws_size,
                              hipStream_t stream) {
    const float* p1   = (const float*)d_in[0];
    const float* p2   = (const float*)d_in[1];
    const float* p3   = (const float*)d_in[2];
    const float* anch = (const float*)d_in[3];
    float* out = (float*)d_out;

    float* w          = (float*)d_ws;
    float* boxes      = w;                                   // B*N*4
    float* planes     = boxes + (size_t)B_IMG * NBOX * 4;    // B*C*N
    float* cand_boxes = planes + (size_t)B_IMG * NCLS * NBOX;// B*C*KC*4
    float* cand_sc    = cand_boxes + (size_t)B_IMG * NCLS * KC * 4; // B*C*KC
    float* fs         = cand_sc + (size_t)B_IMG * NCLS * KC; // B*C*KC

    long long total = (long long)B_IMG * NBOX;
    decode_kernel<<<(unsigned)((total + 255) / 256), 256, 0, stream>>>(
        p1, p2, p3, anch, boxes, planes);
    topk_kernel<<<B_IMG * NCLS, 1024, 0, stream>>>(planes, boxes, cand_boxes, cand_sc);
    nms_kernel<<<B_IMG * NCLS, 32, 0, stream>>>(cand_boxes, cand_sc, fs);
    final_kernel<<<B_IMG, 1024, 0, stream>>>(fs, cand_boxes, out);
}